// TGNNModel_14551349199033
// MI455X (gfx1250) — compile-verified
//
#include <hip/hip_runtime.h>

#define N_NODES 200000
#define N_EDGES 6400000
#define FDIM    128
#define HDIM    16
#define CDIM    10
#define BGRAPH  512
#define NTILES  (N_NODES / 16)   // 12500, exact

typedef float v2f __attribute__((ext_vector_type(2)));
typedef float v8f __attribute__((ext_vector_type(8)));

// ---------------- degree / batch counts ----------------
__global__ void count_deg_kernel(const int* __restrict__ col, int* __restrict__ degcnt) {
    int e = blockIdx.x * blockDim.x + threadIdx.x;
    if (e < N_EDGES) atomicAdd(&degcnt[col[e]], 1);
}

__global__ void node_prep_kernel(const int* __restrict__ degcnt, const int* __restrict__ batch,
                                 float* __restrict__ dis, int* __restrict__ counts) {
    int n = blockIdx.x * blockDim.x + threadIdx.x;
    if (n < N_NODES) {
        dis[n] = rsqrtf(1.0f + (float)degcnt[n]);
        atomicAdd(&counts[batch[n]], 1);
    }
}

// ---------------- layer 1 GEMM: hws = dis * (x @ W1), agg = hws ----------------
// one wave per 16-node tile; V_WMMA_F32_16X16X4_F32, K-loop of 32 steps.
__global__ void __launch_bounds__(256) gemm1_kernel(const float* __restrict__ x,
                                                    const float* __restrict__ W1,
                                                    const float* __restrict__ dis,
                                                    float* __restrict__ hws,
                                                    float* __restrict__ agg) {
    // W1 staged pair-interleaved: lw[(k>>1)*32 + 2n + (k&1)] = W1[k*16+n]
    __shared__ float lw[(FDIM / 2) * 32];
    int tid = threadIdx.x;
#pragma unroll
    for (int i = 0; i < 8; ++i) {
        int idx = i * 256 + tid;          // coalesced read of 128x16
        int k = idx >> 4, n = idx & 15;
        lw[(k >> 1) * 32 + n * 2 + (k & 1)] = W1[idx];
    }
    __syncthreads();

    int wave = tid >> 5, lane = tid & 31;
    int tile = blockIdx.x * 8 + wave;
    if (tile >= NTILES) return;          // wave-uniform: EXEC stays all-ones

    int h  = lane >> 4;                  // lane half selects K+2
    int nn = lane & 15;
    long m = (long)tile * 16 + nn;       // A-matrix row for this lane
    const float* xrow = x + m * FDIM;

    v8f c = {0.f, 0.f, 0.f, 0.f, 0.f, 0.f, 0.f, 0.f};
#pragma unroll
    for (int kk = 0; kk < FDIM; kk += 4) {
        int k0 = kk + 2 * h;
        v2f a = *(const v2f*)(xrow + k0);                 // A: 16x4 fragment
        v2f b = *(const v2f*)(&lw[(k0 >> 1) * 32 + nn * 2]); // B: 4x16 fragment
        c = __builtin_amdgcn_wmma_f32_16x16x4_f32(false, a, false, b,
                                                  (short)0, c, false, false);
    }
    // C layout: vgpr v, lane-half h -> row v + 8h; col = nn
#pragma unroll
    for (int v = 0; v < 8; ++v) {
        long r = (long)tile * 16 + v + 8 * h;
        float val = c[v] * dis[r];       // hws = dis * hw
        hws[r * HDIM + nn] = val;
        agg[r * HDIM + nn] = val;        // self-loop init: agg starts at hws
    }
}

// ---------------- edge scatter: agg[col] += hws[row], per (edge, feature) ----------------
__global__ void edge_scatter_kernel(const int* __restrict__ row, const int* __restrict__ col,
                                    const float* __restrict__ hws, float* __restrict__ agg) {
    long t = (long)blockIdx.x * blockDim.x + threadIdx.x;
    int e = (int)(t >> 4);
    int j = (int)(t & 15);
    if (e < N_EDGES) {
        int r = row[e];
        int ccol = col[e];
        atomicAdd(&agg[(long)ccol * HDIM + j], hws[(long)r * HDIM + j]);
    }
}

// ---------------- layer 2 GEMM with fused layer-1 epilogue ----------------
// input h1[m][k] = relu(dis[m]*agg1[m][k] + b1[k]); output hws2 = dis*(h1@W2), agg2 = hws2
__global__ void __launch_bounds__(256) gemm2_kernel(const float* __restrict__ agg1,
                                                    const float* __restrict__ W2,
                                                    const float* __restrict__ b1v,
                                                    const float* __restrict__ dis,
                                                    float* __restrict__ hws2,
                                                    float* __restrict__ agg2) {
    __shared__ float lw[(HDIM / 2) * 32];
    __shared__ float lb1[HDIM];
    int tid = threadIdx.x;
    if (tid < 256) {
        int k = tid >> 4, n = tid & 15;
        lw[(k >> 1) * 32 + n * 2 + (k & 1)] = W2[tid];
    }
    if (tid < HDIM) lb1[tid] = b1v[tid];
    __syncthreads();

    int wave = tid >> 5, lane = tid & 31;
    int tile = blockIdx.x * 8 + wave;
    if (tile >= NTILES) return;

    int h  = lane >> 4;
    int nn = lane & 15;
    long m = (long)tile * 16 + nn;
    float dm = dis[m];
    const float* arow = agg1 + m * HDIM;

    v8f c = {0.f, 0.f, 0.f, 0.f, 0.f, 0.f, 0.f, 0.f};
#pragma unroll
    for (int kk = 0; kk < HDIM; kk += 4) {
        int k0 = kk + 2 * h;
        v2f raw = *(const v2f*)(arow + k0);
        v2f a;
        a.x = fmaxf(dm * raw.x + lb1[k0], 0.0f);     // fused epilogue of layer 1
        a.y = fmaxf(dm * raw.y + lb1[k0 + 1], 0.0f);
        v2f b = *(const v2f*)(&lw[(k0 >> 1) * 32 + nn * 2]);
        c = __builtin_amdgcn_wmma_f32_16x16x4_f32(false, a, false, b,
                                                  (short)0, c, false, false);
    }
#pragma unroll
    for (int v = 0; v < 8; ++v) {
        long r = (long)tile * 16 + v + 8 * h;
        float val = c[v] * dis[r];
        hws2[r * HDIM + nn] = val;
        agg2[r * HDIM + nn] = val;
    }
}

// ---------------- mean-pool accumulate: pool[batch[n]] += dis[n]*agg2[n] + b2 ----------------
__global__ void pool_kernel(const float* __restrict__ agg2, const float* __restrict__ dis,
                            const float* __restrict__ b2v, const int* __restrict__ batch,
                            float* __restrict__ pool) {
    int t = blockIdx.x * blockDim.x + threadIdx.x;
    int n = t >> 4, j = t & 15;
    if (n < N_NODES) {
        float val = dis[n] * agg2[(long)n * HDIM + j] + b2v[j];  // layer 2: no relu
        atomicAdd(&pool[batch[n] * HDIM + j], val);
    }
}

// ---------------- head: out[b,c] = (pool[b]/count[b]) @ Wfc + bfc ----------------
__global__ void final_kernel(const float* __restrict__ pool, const int* __restrict__ counts,
                             const float* __restrict__ Wfc, const float* __restrict__ bfc,
                             float* __restrict__ out) {
    int t = blockIdx.x * blockDim.x + threadIdx.x;
    if (t < BGRAPH * CDIM) {
        int b = t / CDIM, cc = t % CDIM;
        float inv = 1.0f / fmaxf((float)counts[b], 1.0f);
        float acc = bfc[cc];
#pragma unroll
        for (int j = 0; j < HDIM; ++j)
            acc += pool[b * HDIM + j] * inv * Wfc[j * CDIM + cc];
        out[t] = acc;
    }
}

extern "C" void kernel_launch(void* const* d_in, const int* in_sizes, int n_in,
                              void* d_out, int out_size, void* d_ws, size_t ws_size,
                              hipStream_t stream) {
    (void)in_sizes; (void)n_in; (void)out_size; (void)ws_size;

    const float* x    = (const float*)d_in[0];
    const int*   ei   = (const int*)d_in[1];   // [2, E] flat
    const int*   batch= (const int*)d_in[2];
    const float* W1   = (const float*)d_in[3];
    const float* b1   = (const float*)d_in[4];
    const float* W2   = (const float*)d_in[5];
    const float* b2   = (const float*)d_in[6];
    const float* Wfc  = (const float*)d_in[7];
    const float* bfc  = (const float*)d_in[8];
    float* out = (float*)d_out;

    const int* erow = ei;
    const int* ecol = ei + N_EDGES;

    // workspace carve-out (256B aligned)
    char* ws = (char*)d_ws;
    size_t o = 0;
    auto alloc = [&](size_t bytes) -> void* {
        void* p = ws + o;
        o += (bytes + 255) & ~(size_t)255;
        return p;
    };
    float* dis    = (float*)alloc((size_t)N_NODES * 4);
    int*   degcnt = (int*)  alloc((size_t)N_NODES * 4);
    float* hws    = (float*)alloc((size_t)N_NODES * HDIM * 4);  // reused layer1 & layer2
    float* agg1   = (float*)alloc((size_t)N_NODES * HDIM * 4);
    float* agg2   = (float*)alloc((size_t)N_NODES * HDIM * 4);
    float* pool   = (float*)alloc((size_t)BGRAPH * HDIM * 4);
    int*   counts = (int*)  alloc((size_t)BGRAPH * 4);

    hipMemsetAsync(degcnt, 0, (size_t)N_NODES * 4, stream);
    hipMemsetAsync(counts, 0, (size_t)BGRAPH * 4, stream);
    hipMemsetAsync(pool,   0, (size_t)BGRAPH * HDIM * 4, stream);

    count_deg_kernel<<<(N_EDGES + 255) / 256, 256, 0, stream>>>(ecol, degcnt);
    node_prep_kernel<<<(N_NODES + 255) / 256, 256, 0, stream>>>(degcnt, batch, dis, counts);

    int gemm_grid = (NTILES + 7) / 8;    // 8 waves (tiles) per 256-thread block
    gemm1_kernel<<<gemm_grid, 256, 0, stream>>>(x, W1, dis, hws, agg1);

    int esc_grid = (int)(((long)N_EDGES * 16 + 255) / 256);
    edge_scatter_kernel<<<esc_grid, 256, 0, stream>>>(erow, ecol, hws, agg1);

    gemm2_kernel<<<gemm_grid, 256, 0, stream>>>(agg1, W2, b1, dis, hws, agg2);

    edge_scatter_kernel<<<esc_grid, 256, 0, stream>>>(erow, ecol, hws, agg2);

    pool_kernel<<<(N_NODES * 16 + 255) / 256, 256, 0, stream>>>(agg2, dis, b2, batch, pool);
    final_kernel<<<(BGRAPH * CDIM + 255) / 256, 256, 0, stream>>>(pool, counts, Wfc, bfc, out);
}